// ModulatedDeformConv_26139170964005
// MI455X (gfx1250) — compile-verified
//
#include <hip/hip_runtime.h>
#include <hip/hip_bf16.h>
#include <math.h>

// ---------------------------------------------------------------------------
// ModulatedDeformConv (DCNv2), fp32, B=8 C=64 H=W=128 Cout=64 K=3 pad=1 DG=1
//   Phase 1: offset conv  -> offsets (18 ch) + sigmoid mask (9 ch) in d_ws
//   Phase 2: bilinear-sample V[576 x 16] panel into LDS, then implicit GEMM
//            D[64 x 16] = W[64 x 576] * V[576 x 16] with v_wmma_f32_16x16x4_f32
// ---------------------------------------------------------------------------

typedef __attribute__((ext_vector_type(2))) float v2f;
typedef __attribute__((ext_vector_type(8))) float v8f;

#define HH   128
#define WW   128
#define HW   (HH * WW)       // 16384
#define CIN  64
#define COUT 64
#define KKT  9               // 3x3 taps
#define KDIM (CIN * KKT)     // 576 reduction length
#define TILE_N 16
#define LDSS 17              // padded LDS row stride (floats)
#define NBATCH 8

// ---------------------------------------------------------------------------
// Kernel 1: 27-channel offset/mask conv (3x3, pad 1) + sigmoid on mask chans.
// One thread per output pixel; weights are wave-uniform -> scalar loads.
// ---------------------------------------------------------------------------
__global__ __launch_bounds__(256) void dcn_offset_conv(
    const float* __restrict__ x, const float* __restrict__ ow,
    const float* __restrict__ ob, float* __restrict__ offs,
    float* __restrict__ mask)
{
    int idx = blockIdx.x * 256 + threadIdx.x;     // b*HW + hw  (8*16384 total)
    int b  = idx >> 14;
    int hw = idx & (HW - 1);
    int h  = hw >> 7;
    int w  = hw & (WW - 1);

    const float* xb = x + (size_t)b * CIN * HW;

    float acc[27];
#pragma unroll
    for (int o = 0; o < 27; ++o) acc[o] = ob[o];

    for (int c = 0; c < CIN; ++c) {
        const float* xc = xb + c * HW;
#pragma unroll
        for (int ky = 0; ky < 3; ++ky) {
            int yy = h + ky - 1;
            bool vy = (yy >= 0) && (yy < HH);
#pragma unroll
            for (int kx = 0; kx < 3; ++kx) {
                int xx = w + kx - 1;
                float xv = (vy && xx >= 0 && xx < WW) ? xc[yy * WW + xx] : 0.0f;
                int widx = c * KKT + ky * 3 + kx;   // wave-uniform index
#pragma unroll
                for (int o = 0; o < 27; ++o)
                    acc[o] = fmaf(xv, ow[o * KDIM + widx], acc[o]);
            }
        }
    }

#pragma unroll
    for (int o = 0; o < 18; ++o)
        offs[((size_t)b * 18 + o) * HW + hw] = acc[o];
#pragma unroll
    for (int o = 0; o < 9; ++o)
        mask[((size_t)b * 9 + o) * HW + hw] = 1.0f / (1.0f + __expf(-acc[18 + o]));
}

// ---------------------------------------------------------------------------
// Kernel 2: one block = one batch image b + 16 output pixels.
//   Phase A: 128 threads build V[576][16] (bilinear * mask) in LDS.
//   Phase B: 4 waves, wave m owns Cout rows [16m,16m+16); 144 K-steps of
//            v_wmma_f32_16x16x4_f32 (A from global weight, B from LDS).
// ---------------------------------------------------------------------------
__global__ __launch_bounds__(128) void dcn_deform_gemm(
    const float* __restrict__ x, const float* __restrict__ weight,
    const float* __restrict__ bias, const float* __restrict__ offs,
    const float* __restrict__ mask, float* __restrict__ out)
{
    __shared__ float V[KDIM * LDSS];              // ~38.3 KB

    int tile = blockIdx.x;                        // 0..8191
    int b    = tile >> 10;                        // 1024 tiles per image
    int pix0 = (tile & 1023) * TILE_N;
    int tid  = threadIdx.x;

    const float* xb = x + (size_t)b * CIN * HW;

    // ---------- Phase A: sampling ----------
    for (int pair = tid; pair < TILE_N * KKT; pair += 128) {
        int n  = pair & (TILE_N - 1);
        int kk = pair >> 4;                       // 0..8
        int hw = pix0 + n;
        int h  = hw >> 7;
        int w  = hw & (WW - 1);

        float dy = offs[((size_t)b * 18 + 2 * kk) * HW + hw];
        float dx = offs[((size_t)b * 18 + 2 * kk + 1) * HW + hw];
        float m  = mask[((size_t)b * 9 + kk) * HW + hw];

        float py = dy + (float)(kk / 3) + (float)(h - 1);   // stride 1, pad 1
        float px = dx + (float)(kk % 3) + (float)(w - 1);

        float y0f = floorf(py), x0f = floorf(px);
        float wy = py - y0f,   wx = px - x0f;
        int iy0 = (int)y0f, ix0 = (int)x0f;
        int iy1 = iy0 + 1,  ix1 = ix0 + 1;

        float w00 = (1.0f - wy) * (1.0f - wx) * m;
        float w01 = (1.0f - wy) * wx * m;
        float w10 = wy * (1.0f - wx) * m;
        float w11 = wy * wx * m;

        bool vy0 = (iy0 >= 0) && (iy0 < HH), vy1 = (iy1 >= 0) && (iy1 < HH);
        bool vx0 = (ix0 >= 0) && (ix0 < WW), vx1 = (ix1 >= 0) && (ix1 < WW);
        if (!(vy0 && vx0)) w00 = 0.0f;
        if (!(vy0 && vx1)) w01 = 0.0f;
        if (!(vy1 && vx0)) w10 = 0.0f;
        if (!(vy1 && vx1)) w11 = 0.0f;

        int cy0 = min(max(iy0, 0), HH - 1), cy1 = min(max(iy1, 0), HH - 1);
        int cx0 = min(max(ix0, 0), WW - 1), cx1 = min(max(ix1, 0), WW - 1);
        int i00 = cy0 * WW + cx0, i01 = cy0 * WW + cx1;
        int i10 = cy1 * WW + cx0, i11 = cy1 * WW + cx1;

        for (int c = 0; c < CIN; ++c) {
            const float* xc = xb + c * HW;
            float val = fmaf(w00, xc[i00],
                        fmaf(w01, xc[i01],
                        fmaf(w10, xc[i10],
                             w11 * xc[i11])));
            V[(c * KKT + kk) * LDSS + n] = val;
        }
    }

    __syncthreads();

    // ---------- Phase B: WMMA GEMM ----------
    int wave   = tid >> 5;          // 0..3 -> Cout tile
    int lane   = tid & 31;
    int lane16 = lane & 15;
    int half   = lane >> 4;         // 0: K{k0,k0+1}, 1: K{k0+2,k0+3}

    const float* wrow = weight + (size_t)(wave * 16 + lane16) * KDIM;

    v8f acc = {};
#pragma unroll 4
    for (int k0 = 0; k0 < KDIM; k0 += 4) {
        // A fragment (16x4 fp32): VGPR0 = K = k0+2*half, VGPR1 = K+1
        v2f a;
        a.x = wrow[k0 + 2 * half];
        a.y = wrow[k0 + 2 * half + 1];
        // B fragment (4x16 fp32): row K striped across lanes (N = lane16)
        v2f bf;
        bf.x = V[(k0 + 2 * half) * LDSS + lane16];
        bf.y = V[(k0 + 2 * half + 1) * LDSS + lane16];
        // D = A*B + C   (8 args: neg_a, A, neg_b, B, c_mod, C, reuse_a, reuse_b)
        acc = __builtin_amdgcn_wmma_f32_16x16x4_f32(
            false, a, false, bf, (short)0, acc, false, false);
    }

    // C/D layout: VGPR r, lane l -> M = r + 8*(l>=16), N = l%16
    float* ob = out + ((size_t)b * COUT) * HW + pix0 + lane16;
#pragma unroll
    for (int r = 0; r < 8; ++r) {
        int co = wave * 16 + r + 8 * half;
        ob[(size_t)co * HW] = acc[r] + bias[co];
    }
}

// ---------------------------------------------------------------------------
extern "C" void kernel_launch(void* const* d_in, const int* in_sizes, int n_in,
                              void* d_out, int out_size, void* d_ws, size_t ws_size,
                              hipStream_t stream) {
    const float* x        = (const float*)d_in[0];
    const float* weight   = (const float*)d_in[1];
    const float* bias     = (const float*)d_in[2];
    const float* offset_w = (const float*)d_in[3];
    const float* offset_b = (const float*)d_in[4];
    float* out = (float*)d_out;

    float* offs = (float*)d_ws;                           // 8*18*16384 floats
    float* mask = offs + (size_t)NBATCH * 18 * HW;        // 8* 9*16384 floats

    dcn_offset_conv<<<(NBATCH * HW) / 256, 256, 0, stream>>>(
        x, offset_w, offset_b, offs, mask);

    dcn_deform_gemm<<<NBATCH * (HW / TILE_N), 128, 0, stream>>>(
        x, weight, bias, offs, mask, out);
}